// Terminal_23321672417293
// MI455X (gfx1250) — compile-verified
//
#include <hip/hip_runtime.h>
#include <hip/hip_bf16.h>

typedef __attribute__((ext_vector_type(2))) float v2f;
typedef __attribute__((ext_vector_type(4))) float v4f;
typedef __attribute__((ext_vector_type(8))) float v8f;

#define WG_THREADS 512
#define WG_WAVES   16
#define M_PER_WG   64          // tokens per workgroup (4 M-subtiles of 16)
#define KC         128         // K-chunk staged in LDS
#define A_STRIDE   (KC + 4)    // 132 floats: bank-conflict-free (132 % 64 == 4)
#define L_STRIDE   260         // logits row stride (260 % 64 == 4)
#define NPAD       256         // T=245 padded to 16 N-tiles of 16
#define NTILES     (NPAD / 16)
#define TOPK       8

// ---------------------------------------------------------------------------
// Prologue: swizzle W [D][T] row-major into per-lane WMMA B-fragment order,
// zero-padded to 256 columns:
//   Wp[((ntile * D/4 + k/4) * 32 + lane)] = float2{ W[k+kb][col], W[k+kb+1][col] }
//   where col = ntile*16 + (lane&15), kb = (lane>>4)*2.
// Makes the main-loop B fetch one unconditional, fully-coalesced b64 load.
// ---------------------------------------------------------------------------
__global__ void build_w_swizzle_kernel(const float* __restrict__ W,
                                       v2f* __restrict__ Wp,
                                       int D, int T)
{
    int idx = blockIdx.x * blockDim.x + threadIdx.x;   // float2-entry index
    int kq_per_tile = D >> 2;
    int total = NTILES * kq_per_tile * 32;
    if (idx >= total) return;

    int lane  = idx & 31;
    int kq    = (idx >> 5) % kq_per_tile;
    int ntile = (idx >> 5) / kq_per_tile;
    int col   = ntile * 16 + (lane & 15);
    int k0    = kq * 4 + ((lane >> 4) << 1);

    v2f v;
    v[0] = (col < T) ? W[(size_t)k0 * T + col]       : 0.0f;
    v[1] = (col < T) ? W[(size_t)(k0 + 1) * T + col] : 0.0f;
    Wp[idx] = v;
}

// ---------------------------------------------------------------------------
// Main fused kernel: x passthrough copy + WMMA router GEMM + softmax + top-8.
// ---------------------------------------------------------------------------
__global__ __launch_bounds__(WG_THREADS, 2)
void terminal_router_topk_kernel(const float* __restrict__ x,
                                 const float* __restrict__ Wp,     // swizzled B frags
                                 const float* __restrict__ bias,   // [T]
                                 const int*   __restrict__ conn,   // [T][3]
                                 float* __restrict__ out_x,
                                 float* __restrict__ out_p,
                                 float* __restrict__ out_sel,
                                 int D, int T)
{
    __shared__ float smem[M_PER_WG * L_STRIDE];   // 66,560 B; A-chunk then logits

    const int tid  = threadIdx.x;
    const int wave = tid >> 5;
    const int lane = tid & 31;
    const int lrow = lane & 15;          // M position within fragment
    const int kb   = (lane >> 4) << 1;   // lanes 0-15: K{0,1}; lanes 16-31: K{2,3}
    const int tok0 = blockIdx.x * M_PER_WG;
    const int n0   = wave * 16;          // this wave's N-tile base

    // per-wave B fragment stream: contiguous float2 per (k-group, lane)
    const v2f* __restrict__ bwave =
        (const v2f*)Wp + ((size_t)wave * (D >> 2)) * 32 + lane;

    v8f zero = {};
    v8f acc0 = zero, acc1 = zero, acc2 = zero, acc3 = zero;

    for (int kc = 0; kc < D; kc += KC) {
        // ---- cooperative A-chunk load into LDS, fused with x passthrough ----
        #pragma unroll
        for (int i = 0; i < (M_PER_WG * KC) / (WG_THREADS * 4); ++i) {
            int f  = (tid + i * WG_THREADS) << 2;      // flat float index in chunk
            int t  = f >> 7;                           // / KC
            int k  = f & (KC - 1);
            size_t gidx = (size_t)(tok0 + t) * D + kc + k;
            v4f v = __builtin_nontemporal_load((const v4f*)(x + gidx));
            *(v4f*)(&smem[t * A_STRIDE + k]) = v;
            __builtin_nontemporal_store(v, (v4f*)(out_x + gidx));
        }
        __syncthreads();

        const v2f* __restrict__ bchunk = bwave + (size_t)(kc >> 2) * 32;

        // ---- WMMA accumulate: 1 coalesced B load feeds 4 M-subtiles ----
        #pragma unroll 4
        for (int kk = 0; kk < KC; kk += 4) {
            v2f b = bchunk[(kk >> 2) * 32];
            const float* abase = &smem[lrow * A_STRIDE + kk + kb];
            v2f a0 = *(const v2f*)(abase + 0 * 16 * A_STRIDE);
            v2f a1 = *(const v2f*)(abase + 1 * 16 * A_STRIDE);
            v2f a2 = *(const v2f*)(abase + 2 * 16 * A_STRIDE);
            v2f a3 = *(const v2f*)(abase + 3 * 16 * A_STRIDE);
            acc0 = __builtin_amdgcn_wmma_f32_16x16x4_f32(false, a0, false, b, (short)0, acc0, false, false);
            acc1 = __builtin_amdgcn_wmma_f32_16x16x4_f32(false, a1, false, b, (short)0, acc1, false, false);
            acc2 = __builtin_amdgcn_wmma_f32_16x16x4_f32(false, a2, false, b, (short)0, acc2, false, false);
            acc3 = __builtin_amdgcn_wmma_f32_16x16x4_f32(false, a3, false, b, (short)0, acc3, false, false);
        }
        __syncthreads();
    }

    // ---- dump logits to LDS (C layout: lanes0-15 row r, lanes16-31 row r+8) ----
    {
        int mrow = (lane >> 4) << 3;     // 0 or 8
        int n = n0 + lrow;
        #pragma unroll
        for (int r = 0; r < 8; ++r) {
            smem[(0 * 16 + mrow + r) * L_STRIDE + n] = acc0[r];
            smem[(1 * 16 + mrow + r) * L_STRIDE + n] = acc1[r];
            smem[(2 * 16 + mrow + r) * L_STRIDE + n] = acc2[r];
            smem[(3 * 16 + mrow + r) * L_STRIDE + n] = acc3[r];
        }
    }
    __syncthreads();

    // ---- softmax + top-8 per token; each wave owns 4 tokens ----
    const float NEG_INF = -__builtin_inff();
    for (int i = 0; i < M_PER_WG / WG_WAVES; ++i) {
        int tl = wave * (M_PER_WG / WG_WAVES) + i;
        size_t tg = (size_t)(tok0 + tl);

        float v[NPAD / 32];
        #pragma unroll
        for (int j = 0; j < NPAD / 32; ++j) {
            int c = lane + 32 * j;
            v[j] = (c < T) ? (smem[tl * L_STRIDE + c] + bias[c]) : NEG_INF;
        }

        // wave-wide max
        float mx = v[0];
        #pragma unroll
        for (int j = 1; j < NPAD / 32; ++j) mx = fmaxf(mx, v[j]);
        for (int off = 16; off > 0; off >>= 1) mx = fmaxf(mx, __shfl_xor(mx, off, 32));

        // wave-wide sum of exp
        float s = 0.0f;
        #pragma unroll
        for (int j = 0; j < NPAD / 32; ++j)
            s += (v[j] == NEG_INF) ? 0.0f : expf(v[j] - mx);
        for (int off = 16; off > 0; off >>= 1) s += __shfl_xor(s, off, 32);
        float inv = 1.0f / s;

        // extract top-8 (descending; ties -> lowest index, like jax.lax.top_k)
        for (int sel = 0; sel < TOPK; ++sel) {
            float bv = v[0]; int bj = 0;
            #pragma unroll
            for (int j = 1; j < NPAD / 32; ++j)
                if (v[j] > bv) { bv = v[j]; bj = j; }
            float rv = bv;
            int   rc = lane + 32 * bj;
            for (int off = 16; off > 0; off >>= 1) {
                float ov = __shfl_xor(rv, off, 32);
                int   oc = __shfl_xor(rc, off, 32);
                if (ov > rv || (ov == rv && oc < rc)) { rv = ov; rc = oc; }
            }
            if ((rc & 31) == lane) v[rc >> 5] = NEG_INF;   // owner retires winner
            if (lane == 0) {
                out_p[tg * TOPK + sel] = expf(rv - mx) * inv;
                size_t so = (tg * TOPK + sel) * 3;
                out_sel[so + 0] = (float)conn[rc * 3 + 0];
                out_sel[so + 1] = (float)conn[rc * 3 + 1];
                out_sel[so + 2] = (float)conn[rc * 3 + 2];
            }
        }
    }
}

extern "C" void kernel_launch(void* const* d_in, const int* in_sizes, int n_in,
                              void* d_out, int out_size, void* d_ws, size_t ws_size,
                              hipStream_t stream)
{
    const float* x    = (const float*)d_in[0];
    const float* W    = (const float*)d_in[1];
    const float* bias = (const float*)d_in[2];
    const int*   conn = (const int*)d_in[3];
    // d_in[4] = top_k scalar on device; reference constant TOPK=8 baked in.

    int T = in_sizes[2];                          // 245
    int D = in_sizes[1] / T;                      // 2048
    long long ntok = (long long)in_sizes[0] / D;  // B*S = 32768

    float* out_x   = (float*)d_out;
    float* out_p   = out_x + (size_t)ntok * D;
    float* out_sel = out_p + (size_t)ntok * TOPK;

    v2f* Wp = (v2f*)d_ws;                         // NTILES * (D/4) * 32 float2 = 2 MB

    int wp_entries = NTILES * (D >> 2) * 32;
    build_w_swizzle_kernel<<<(wp_entries + 255) / 256, 256, 0, stream>>>(W, Wp, D, T);

    dim3 grid((unsigned)(ntok / M_PER_WG));       // 512 workgroups
    terminal_router_topk_kernel<<<grid, WG_THREADS, 0, stream>>>(
        x, (const float*)Wp, bias, conn, out_x, out_p, out_sel, D, T);
}